// IntensityEstimator_48241072669055
// MI455X (gfx1250) — compile-verified
//
#include <hip/hip_runtime.h>
#include <math.h>

#define NSPOTS_PB 16              // spots per block
#define FITS_PB   (NSPOTS_PB * 6) // 96 threads = 3 waves
#define RR        16
#define NPX       256
#define STRIDE    258             // padded LDS row stride (floats) -> conflict-free banking
#define ITERS     50
#define LAMBDA_   100.0f

typedef float        v2f   __attribute__((ext_vector_type(2)));
typedef float        v8f   __attribute__((ext_vector_type(8)));
typedef unsigned int u32x4 __attribute__((ext_vector_type(4)));
typedef int          i32x4 __attribute__((ext_vector_type(4)));
typedef int          i32x8 __attribute__((ext_vector_type(8)));

__global__ __launch_bounds__(FITS_PB)
void intensity_fit_kernel(const float* __restrict__ params,
                          const float* __restrict__ data,
                          float* __restrict__ out,
                          int nspots)
{
    __shared__ __align__(16) float s_smp [FITS_PB   * STRIDE]; //  99072 B
    __shared__ __align__(16) float s_psf [NSPOTS_PB * STRIDE]; //  16512 B
    __shared__ __align__(16) float s_psf2[NSPOTS_PB * STRIDE]; //  16512 B
    __shared__ float s_edge[2][NSPOTS_PB][RR];                 //   2048 B

    const int t     = threadIdx.x;
    const int lane  = t & 31;
    const int wv    = t >> 5;
    const int spot0 = blockIdx.x * NSPOTS_PB;
    const int fit0  = blockIdx.x * FITS_PB;
    const int nfits = nspots * 6;
    const int rowsLocal  = min(FITS_PB,   nfits  - fit0);
    const int spotsLocal = min(NSPOTS_PB, nspots - spot0);

    // ---- Phase 0: TDM DMA of this block's ROI tile (96 x 256 f32) into LDS.
    // Hardware pads 2 DWORDs after every 256 DWORDs -> LDS row stride 258.
    // Issued by wave 0 only (scalar branch via readfirstlane; TDM ignores EXEC).
    if (__builtin_amdgcn_readfirstlane(wv) == 0) {
        unsigned long long gaddr = (unsigned long long)(uintptr_t)data
                                 + (unsigned long long)fit0 * (NPX * 4ull);
        unsigned ldsOff = (unsigned)(uintptr_t)&s_smp[0]; // low 32 bits = LDS byte offset
        u32x4 g0;
        g0[0] = 1u;                                        // count=1, user descriptor
        g0[1] = ldsOff;                                    // lds_addr
        g0[2] = (unsigned)(gaddr & 0xffffffffull);         // global_addr[31:0]
        g0[3] = (unsigned)((gaddr >> 32) & 0x1ffffffull)   // global_addr[56:32]
              | 0x80000000u;                               // type=2 ("image")
        i32x8 g1;
        g1[0] = (int)((2u << 16)     // data_size = 4B
                    | (1u << 20)     // pad_enable
                    | (7u << 22)     // pad_interval: every 256 DWORDs
                    | (1u << 25));   // pad_amount: 2 DWORDs
        g1[1] = (int)(((unsigned)NPX & 0xffffu) << 16);        // tensor_dim0 lo16
        g1[2] = (int)(((unsigned)rowsLocal & 0xffffu) << 16);  // dim0 hi=0 | tensor_dim1 lo16
        g1[3] = (int)(((unsigned)NPX) << 16);                  // dim1 hi=0 | tile_dim0=256
        g1[4] = (int)((unsigned)rowsLocal & 0xffffu);          // tile_dim1 | tile_dim2=0
        g1[5] = (int)NPX;                                      // tensor_dim0_stride lo32
        g1[6] = 0;
        g1[7] = 0;
        i32x4 gz4 = {0, 0, 0, 0};
        i32x8 gz8 = {0, 0, 0, 0, 0, 0, 0, 0};
        __builtin_amdgcn_tensor_load_to_lds(g0, g1, gz4, gz4, gz8, 0);
    }

    // ---- Phase 1: erf edge tables (overlaps with the TDM DMA).
    const float cinv = 0.47140452079103173f; // 1/(sqrt(2)*1.5)
    for (int e = t; e < 2 * NSPOTS_PB * RR; e += FITS_PB) {
        int s    = e >> 5;
        int axis = (e >> 4) & 1;   // 0 -> Ex (x-center), 1 -> Ey (y-center)
        int pix  = e & 15;
        float center = (s < spotsLocal) ? params[(spot0 + s) * 5 + axis] : 8.0f;
        float d = (float)pix - center;
        s_edge[axis][s][pix] = 0.5f * (erff((d + 1.0f) * cinv) - erff(d * cinv));
    }
    __syncthreads();

    // ---- Phase 2: PSF = Ey (x) Ex via rank-1 WMMA per spot (wave-level, EXEC all-1s).
    for (int s = wv; s < NSPOTS_PB; s += 3) {
        float eyv = s_edge[1][s][lane & 15];
        float exv = s_edge[0][s][lane & 15];
        v2f a = {0.0f, 0.0f}, b = {0.0f, 0.0f};
        a[0] = (lane < 16) ? eyv : 0.0f;   // A(16x4): col K=0 = Ey, rest 0
        b[0] = (lane < 16) ? exv : 0.0f;   // B(4x16): row K=0 = Ex, rest 0
        v8f c = {0, 0, 0, 0, 0, 0, 0, 0};
        v8f d = __builtin_amdgcn_wmma_f32_16x16x4_f32(
                    false, a, false, b, (short)0, c, false, false);
        int mb = (lane < 16) ? 0 : 8;      // C/D layout: VGPR i -> rows i and i+8
        int n  = lane & 15;
        #pragma unroll
        for (int i = 0; i < 8; ++i) {
            float pv = d[i];
            int px = (mb + i) * RR + n;
            s_psf [s * STRIDE + px] = pv;
            s_psf2[s * STRIDE + px] = pv * pv;
        }
    }

    __builtin_amdgcn_s_wait_tensorcnt(0);  // wave0 drains its TDM; others pass
    __syncthreads();                       // ROI + PSF now visible block-wide

    // ---- Phase 3: 50 Newton iterations, thread-per-fit, all-LDS-resident.
    if (t < rowsLocal) {
        const int sl = t / 6;
        const float* pS  = &s_smp [t  * STRIDE];
        const float* pP  = &s_psf [sl * STRIDE];
        const float* pP2 = &s_psf2[sl * STRIDE];

        float I  = params[(spot0 + sl) * 5 + 4] * (1.0f / 6.0f);
        float bg = 0.0f;

        for (int it = 0; it < ITERS; ++it) {
            v2f a0 = {0,0}, a1 = {0,0}, a2 = {0,0}, r0 = {0,0}, r1 = {0,0};
            v2f Iv = {I, I}, bgv = {bg, bg};
            #pragma unroll 4
            for (int px = 0; px < NPX; px += 2) {
                v2f pv  = *(const v2f*)(pP  + px);
                v2f pv2 = *(const v2f*)(pP2 + px);
                v2f sm  = *(const v2f*)(pS  + px);
                v2f mu  = __builtin_elementwise_fma(Iv, pv, bgv);
                v2f inv;
                inv[0] = __builtin_amdgcn_rcpf(fmaxf(mu[0], 1e-9f));
                inv[1] = __builtin_amdgcn_rcpf(fmaxf(mu[1], 1e-9f));
                v2f tt = sm * inv;
                v2f af = tt * inv;
                v2f df = tt - (v2f){1.0f, 1.0f};
                a0 = __builtin_elementwise_fma(af, pv2, a0);
                a1 = __builtin_elementwise_fma(af, pv,  a1);
                a2 += af;
                r0 = __builtin_elementwise_fma(df, pv,  r0);
                r1 += df;
            }
            float A00 = a0[0] + a0[1] + LAMBDA_;
            float A01 = a1[0] + a1[1];
            float A11 = a2[0] + a2[1] + LAMBDA_;
            float B0  = r0[0] + r0[1];
            float B1  = r1[0] + r1[1];
            float det  = A00 * A11 - A01 * A01;
            float rdet = __builtin_amdgcn_rcpf(det);
            I  = fminf(fmaxf(I  + (A11 * B0 - A01 * B1) * rdet, 1.0f), 1.0e6f);
            bg = fminf(fmaxf(bg + (A00 * B1 - A01 * B0) * rdet, 1.0f), 1.0e3f);
        }

        // ---- Phase 4: CRLB + chi-square.
        float F00 = 0.f, F01 = 0.f, F11 = 0.f, chis = 0.f;
        #pragma unroll 4
        for (int px = 0; px < NPX; ++px) {
            float pv = pP[px], pv2 = pP2[px], sm = pS[px];
            float mu  = fmaf(I, pv, bg);
            float inv = __builtin_amdgcn_rcpf(fmaxf(mu, 1e-9f));
            F00 = fmaf(pv2, inv, F00);
            F01 = fmaf(pv,  inv, F01);
            F11 += inv;
            float r = sm - mu;
            chis = fmaf(r * r, __builtin_amdgcn_rcpf(mu + 0.01f), chis);
        }
        float detF  = F00 * F11 - F01 * F01;
        float rdetF = 1.0f / detF;
        int f = fit0 + t;
        out[f * 5 + 0] = I;
        out[f * 5 + 1] = bg;
        out[f * 5 + 2] = sqrtf(F11 * rdetF);
        out[f * 5 + 3] = sqrtf(F00 * rdetF);
        out[f * 5 + 4] = chis;
    }
}

extern "C" void kernel_launch(void* const* d_in, const int* in_sizes, int n_in,
                              void* d_out, int out_size, void* d_ws, size_t ws_size,
                              hipStream_t stream)
{
    const float* params = (const float*)d_in[0];
    const float* data   = (const float*)d_in[1];
    float* out = (float*)d_out;
    int nspots = in_sizes[0] / 5;                      // params is [nspots, 5]
    int blocks = (nspots + NSPOTS_PB - 1) / NSPOTS_PB; // 1250 for nspots=20000
    intensity_fit_kernel<<<blocks, FITS_PB, 0, stream>>>(params, data, out, nspots);
}